// LocalAttention_8211977470525
// MI455X (gfx1250) — compile-verified
//
#include <hip/hip_runtime.h>
#include <stdint.h>

// ---- problem constants (from reference) ----
#define N_NODES   50000
#define N_EDGES   1600000
#define DIM       128
#define HEADS     4
#define HEAD_DIM  32
#define NEG_SLOPE 0.2f
#define E_TOT     (N_EDGES + N_NODES)   // edges + self loops

typedef __attribute__((ext_vector_type(2))) float v2f;
typedef __attribute__((ext_vector_type(4))) float v4f;
typedef __attribute__((ext_vector_type(8))) float v8f;

// ---------------------------------------------------------------------------
// monotone float <-> uint32 key so float segment-max can use HW atomicMax(u32)
// ---------------------------------------------------------------------------
__device__ __forceinline__ unsigned fkey(float f) {
  unsigned u = __float_as_uint(f);
  return (u & 0x80000000u) ? ~u : (u | 0x80000000u);
}
__device__ __forceinline__ float funkey(unsigned k) {
  return (k & 0x80000000u) ? __uint_as_float(k ^ 0x80000000u)
                           : __uint_as_float(~k);
}

__device__ __forceinline__ float leaky(float v) {
  return v >= 0.0f ? v : NEG_SLOPE * v;
}

// ---------------------------------------------------------------------------
// init: zero out (poisoned by harness), sum accumulators, and max keys
// (key 0 == minus-infinity under the monotone map; self loops guarantee
//  every destination/head gets at least one real update)
// ---------------------------------------------------------------------------
__global__ void gat_init(float* __restrict__ out, unsigned* __restrict__ mkey,
                         float* __restrict__ ssum) {
  int i = blockIdx.x * blockDim.x + threadIdx.x;
  if (i < N_NODES * DIM) out[i] = 0.0f;
  if (i < N_NODES * HEADS) { mkey[i] = 0u; ssum[i] = 0.0f; }
}

// ---------------------------------------------------------------------------
// h = x @ W^T via V_WMMA_F32_16X16X4_F32.
// One 16x16 output tile per wave; block = 8 waves = one 16-row slab x all
// 8 column tiles. A-slab staged in LDS (pitch 132 floats -> conflict-free).
//
// f32 WMMA VGPR layouts (ISA 7.12.2):
//   A 16x4 : vgpr j, lane L -> M = L%16, K = 2*(L/16) + j
//   B 4x16 : vgpr j, lane L -> N = L%16, K = 2*(L/16) + j   (B[k][n] = W[n][k])
//   C/D    : vgpr r, lane L -> M = r + 8*(L/16), N = L%16
// ---------------------------------------------------------------------------
#define A_PITCH 132

__global__ __launch_bounds__(256) void gat_gemm(const float* __restrict__ x,
                                                const float* __restrict__ W,
                                                float* __restrict__ h) {
  __shared__ float As[16 * A_PITCH];           // 8.25 KB of the 320 KB WGP LDS
  const int row0 = blockIdx.x * 16;            // 50000 / 16 = 3125 exact
  const int tid  = threadIdx.x;

  // cooperative load: 16 rows x 128 cols of x
  for (int i = tid; i < 16 * DIM; i += 256) {
    int r = i >> 7, c = i & 127;
    As[r * A_PITCH + c] = x[(size_t)(row0 + r) * DIM + c];
  }
  __syncthreads();

  const int wave = tid >> 5;                   // column tile 0..7
  const int lane = tid & 31;
  const int n0   = wave * 16;
  const int lm   = lane & 15;
  const int lh   = lane >> 4;                  // 0 or 1

  v8f acc = {};
  const float* wrow = W + (size_t)(n0 + lm) * DIM;   // row of W for this lane
  const float* arow = As + lm * A_PITCH;

#pragma unroll 8
  for (int k0 = 0; k0 < DIM; k0 += 4) {
    const int kk = k0 + 2 * lh;
    v2f a, b;
    a.x = arow[kk + 0];
    a.y = arow[kk + 1];
    b.x = wrow[kk + 0];
    b.y = wrow[kk + 1];
    acc = __builtin_amdgcn_wmma_f32_16x16x4_f32(
        /*neg_a=*/false, a, /*neg_b=*/false, b,
        /*c_mod=*/(short)0, acc, /*reuse_a=*/false, /*reuse_b=*/false);
  }

#pragma unroll
  for (int r = 0; r < 8; ++r)
    h[(size_t)(row0 + r + 8 * lh) * DIM + n0 + lm] = acc[r];
}

// ---------------------------------------------------------------------------
// per-node attention halves: a_src[n,h] = <h[n,h,:], att_src[h,:]>  (and dst)
// ---------------------------------------------------------------------------
__global__ void gat_avec(const float* __restrict__ h,
                         const float* __restrict__ att_src,
                         const float* __restrict__ att_dst,
                         float* __restrict__ a_src, float* __restrict__ a_dst) {
  int i = blockIdx.x * blockDim.x + threadIdx.x;   // over N*HEADS
  if (i >= N_NODES * HEADS) return;
  int n = i >> 2, hd = i & 3;
  const float* hp = h + (size_t)n * DIM + hd * HEAD_DIM;
  const float* as = att_src + hd * HEAD_DIM;
  const float* ad = att_dst + hd * HEAD_DIM;
  float ss = 0.0f, dd = 0.0f;
#pragma unroll 8
  for (int c = 0; c < HEAD_DIM; ++c) {
    float v = hp[c];
    ss = fmaf(v, as[c], ss);
    dd = fmaf(v, ad[c], dd);
  }
  a_src[i] = ss;
  a_dst[i] = dd;
}

// ---------------------------------------------------------------------------
// edge pass 1: segment max of leaky-relu logits, grouped by dst
// ---------------------------------------------------------------------------
__global__ void gat_edge_max(const long long* __restrict__ ei,
                             const float* __restrict__ a_src,
                             const float* __restrict__ a_dst,
                             unsigned* __restrict__ mkey) {
  int e = blockIdx.x * blockDim.x + threadIdx.x;
  if (e >= E_TOT) return;
  int s, d;
  if (e < N_EDGES) { s = (int)ei[e]; d = (int)ei[N_EDGES + e]; }
  else             { s = d = e - N_EDGES; }
#pragma unroll
  for (int hd = 0; hd < HEADS; ++hd) {
    float l = leaky(a_src[s * HEADS + hd] + a_dst[d * HEADS + hd]);
    atomicMax(&mkey[d * HEADS + hd], fkey(l));
  }
}

// ---------------------------------------------------------------------------
// edge pass 2: segment sum of exp(logit - max)
// ---------------------------------------------------------------------------
__global__ void gat_edge_sum(const long long* __restrict__ ei,
                             const float* __restrict__ a_src,
                             const float* __restrict__ a_dst,
                             const unsigned* __restrict__ mkey,
                             float* __restrict__ ssum) {
  int e = blockIdx.x * blockDim.x + threadIdx.x;
  if (e >= E_TOT) return;
  int s, d;
  if (e < N_EDGES) { s = (int)ei[e]; d = (int)ei[N_EDGES + e]; }
  else             { s = d = e - N_EDGES; }
#pragma unroll
  for (int hd = 0; hd < HEADS; ++hd) {
    float l = leaky(a_src[s * HEADS + hd] + a_dst[d * HEADS + hd]);
    float m = funkey(mkey[d * HEADS + hd]);
    unsafeAtomicAdd(&ssum[d * HEADS + hd], __expf(l - m));
  }
}

// ---------------------------------------------------------------------------
// edge pass 3: out[dst] += h[src] * alpha.  One wave32 per edge; each lane
// moves float4 (coalesced 512B gather + 4 HW f32 atomic adds per lane).
// head of column block = lane/8.
// ---------------------------------------------------------------------------
__global__ __launch_bounds__(256) void gat_edge_aggr(
    const long long* __restrict__ ei, const float* __restrict__ h,
    const float* __restrict__ a_src, const float* __restrict__ a_dst,
    const unsigned* __restrict__ mkey, const float* __restrict__ ssum,
    float* __restrict__ out) {
  int e    = (blockIdx.x * blockDim.x + threadIdx.x) >> 5;
  int lane = threadIdx.x & 31;
  if (e >= E_TOT) return;
  int s, d;
  if (e < N_EDGES) { s = (int)ei[e]; d = (int)ei[N_EDGES + e]; }
  else             { s = d = e - N_EDGES; }

  int hd  = lane >> 3;                     // 8 lanes (32 floats) per head
  float l = leaky(a_src[s * HEADS + hd] + a_dst[d * HEADS + hd]);
  float m = funkey(mkey[d * HEADS + hd]);
  float alpha = __expf(l - m) / (ssum[d * HEADS + hd] + 1e-16f);

  const v4f hv = *(const v4f*)(h + (size_t)s * DIM + lane * 4);
  float* op = out + (size_t)d * DIM + lane * 4;
  unsafeAtomicAdd(op + 0, hv.x * alpha);
  unsafeAtomicAdd(op + 1, hv.y * alpha);
  unsafeAtomicAdd(op + 2, hv.z * alpha);
  unsafeAtomicAdd(op + 3, hv.w * alpha);
}

// ---------------------------------------------------------------------------
// launch
// ---------------------------------------------------------------------------
extern "C" void kernel_launch(void* const* d_in, const int* in_sizes, int n_in,
                              void* d_out, int out_size, void* d_ws, size_t ws_size,
                              hipStream_t stream) {
  const float*     x       = (const float*)d_in[0];       // [N, 128]
  const float*     W       = (const float*)d_in[1];       // [128, 128]
  const float*     att_src = (const float*)d_in[2];       // [4, 32]
  const float*     att_dst = (const float*)d_in[3];       // [4, 32]
  const long long* ei      = (const long long*)d_in[4];   // [2, E] int64
  float*           out     = (float*)d_out;               // [N, 128]

  // workspace layout (28.8 MB total): h | a_src | a_dst | mkey | ssum
  float*    h     = (float*)d_ws;
  float*    a_src = h + (size_t)N_NODES * DIM;
  float*    a_dst = a_src + (size_t)N_NODES * HEADS;
  unsigned* mkey  = (unsigned*)(a_dst + (size_t)N_NODES * HEADS);
  float*    ssum  = (float*)(mkey + (size_t)N_NODES * HEADS);

  // init out / max keys / sums
  {
    int total = N_NODES * DIM;
    gat_init<<<(total + 255) / 256, 256, 0, stream>>>(out, mkey, ssum);
  }
  // GEMM: h = x @ W^T   (3125 blocks x 8 waves, 16x16 WMMA tiles)
  gat_gemm<<<N_NODES / 16, 256, 0, stream>>>(x, W, h);
  // per-node attention dots
  {
    int total = N_NODES * HEADS;
    gat_avec<<<(total + 255) / 256, 256, 0, stream>>>(h, att_src, att_dst, a_src, a_dst);
  }
  // edge passes
  {
    int blocks = (E_TOT + 255) / 256;
    gat_edge_max<<<blocks, 256, 0, stream>>>(ei, a_src, a_dst, mkey);
    gat_edge_sum<<<blocks, 256, 0, stream>>>(ei, a_src, a_dst, mkey, ssum);
  }
  {
    long long threads = (long long)E_TOT * 32;
    int blocks = (int)((threads + 255) / 256);
    gat_edge_aggr<<<blocks, 256, 0, stream>>>(ei, h, a_src, a_dst, mkey, ssum, out);
  }
}